// Transfer_35399120453953
// MI455X (gfx1250) — compile-verified
//
#include <hip/hip_runtime.h>

typedef __attribute__((ext_vector_type(2))) float v2f;
typedef __attribute__((ext_vector_type(8))) float v8f;

#define C_IN  64
#define C_OUT 128

// ---------------------------------------------------------------------------
// Kernel A: fused gather + segment-sum.  One wave32 per domain.
// domain_ids is sorted, so each wave binary-searches its [start,end) range.
// Lane l owns channels 2l and 2l+1  ->  each entry is a coalesced 256B row read.
// ---------------------------------------------------------------------------
__global__ __launch_bounds__(256) void seg_sum_kernel(
    const float* __restrict__ features,      // [N_ATOMS, 64]
    const int*   __restrict__ atom_indices,  // [E]
    const int*   __restrict__ domain_ids,    // [E], sorted
    float*       __restrict__ transferred,   // [M, 64]  (workspace)
    int num_entries, int num_domains)
{
  const int wave = threadIdx.x >> 5;
  const int lane = threadIdx.x & 31;
  const long d = (long)blockIdx.x * 8 + wave;
  if (d >= num_domains) return;  // wave-uniform

  // lower_bound(d)
  int lo = 0, hi = num_entries;
  while (lo < hi) {
    int mid = (lo + hi) >> 1;
    if (domain_ids[mid] < (int)d) lo = mid + 1; else hi = mid;
  }
  const int start = lo;
  // lower_bound(d+1), starting from `start`
  hi = num_entries;
  while (lo < hi) {
    int mid = (lo + hi) >> 1;
    if (domain_ids[mid] < (int)d + 1) lo = mid + 1; else hi = mid;
  }
  const int end = lo;

  v2f acc = {0.f, 0.f};
  for (int e = start; e < end; ++e) {
    const int a = atom_indices[e];                       // wave-uniform load
    const v2f r = *(const v2f*)(features + (long)a * C_IN + 2 * lane);
    acc.x += r.x;
    acc.y += r.y;
  }
  *(v2f*)(transferred + d * C_IN + 2 * lane) = acc;
}

// ---------------------------------------------------------------------------
// Kernel B: [M,64] x [64,128] + bias via V_WMMA_F32_16X16X4_F32 (full f32).
// One wave32 per 16x16 output tile; K=64 -> 16 WMMA steps of K=4.
//
// ISA fragment layouts (wave32):
//   A 16x4 : lanes 0-15 -> M=lane,    VGPR0=K0, VGPR1=K1
//            lanes16-31 -> M=lane-16, VGPR0=K2, VGPR1=K3
//   B 4x16 : lanes 0-15 -> N=lane,    VGPR0=K0, VGPR1=K1
//            lanes16-31 -> N=lane-16, VGPR0=K2, VGPR1=K3
//   C/D    : VGPR v: lanes 0-15 -> M=v, lanes 16-31 -> M=v+8; N=lane&15
// ---------------------------------------------------------------------------
__global__ __launch_bounds__(256) void gemm_wmma_kernel(
    const float* __restrict__ A,    // [M, 64]   (transferred)
    const float* __restrict__ Wm,   // [64, 128]
    const float* __restrict__ bias, // [128]
    float*       __restrict__ out,  // [M, 128]
    int M)
{
  const int wave = threadIdx.x >> 5;
  const int lane = threadIdx.x & 31;
  const int numRowTiles = (M + 15) >> 4;
  const long numTiles = (long)numRowTiles * (C_OUT / 16);

  const long tile = (long)blockIdx.x * 8 + wave;
  if (tile >= numTiles) return;  // wave-uniform -> EXEC full in WMMA path

  const int rowTile = (int)(tile >> 3);   // C_OUT/16 == 8 col tiles
  const int colTile = (int)(tile & 7);
  const int m0 = rowTile << 4;
  const int n0 = colTile << 4;

  const int half = lane >> 4;     // 0: K pair {0,1}, 1: K pair {2,3}
  const int l15  = lane & 15;

  const float* Arow = A + (long)(m0 + l15) * C_IN;

  v8f acc = {};
#pragma unroll
  for (int k0 = 0; k0 < C_IN; k0 += 4) {
    const int ka = k0 + 2 * half;
    // A fragment: 2 consecutive K values of row (m0+l15)
    v2f afrag = *(const v2f*)(Arow + ka);
    // B fragment: W rows ka, ka+1 at column n0+l15
    v2f bfrag;
    bfrag.x = Wm[(ka + 0) * C_OUT + n0 + l15];
    bfrag.y = Wm[(ka + 1) * C_OUT + n0 + l15];
    // (neg_a, A, neg_b, B, c_mod, C, reuse_a, reuse_b)
    acc = __builtin_amdgcn_wmma_f32_16x16x4_f32(
        false, afrag, false, bfrag, (short)0, acc, false, false);
  }

  const float bv = bias[n0 + l15];
#pragma unroll
  for (int v = 0; v < 8; ++v) {
    const int m = m0 + v + 8 * half;
    if (m < M) out[(long)m * C_OUT + n0 + l15] = acc[v] + bv;
  }
}

// ---------------------------------------------------------------------------
extern "C" void kernel_launch(void* const* d_in, const int* in_sizes, int n_in,
                              void* d_out, int out_size, void* d_ws, size_t ws_size,
                              hipStream_t stream)
{
  const float* features     = (const float*)d_in[0];
  const int*   atom_indices = (const int*)  d_in[1];
  const int*   domain_ids   = (const int*)  d_in[2];
  // d_in[3] is the num_domains scalar on DEVICE; derive M on host instead:
  const float* Wm   = (const float*)d_in[4];
  const float* bias = (const float*)d_in[5];

  const int num_entries = in_sizes[1];
  const int M = out_size / C_OUT;          // == num_domains

  float* transferred = (float*)d_ws;       // needs M*64*4 bytes (64 MB)
  float* out = (float*)d_out;

  // Kernel A: 8 waves (domains) per 256-thread block
  {
    int blocks = (M + 7) / 8;
    seg_sum_kernel<<<blocks, 256, 0, stream>>>(
        features, atom_indices, domain_ids, transferred, num_entries, M);
  }

  // Kernel B: 8 waves (output tiles) per 256-thread block
  {
    long numTiles = (long)((M + 15) / 16) * (C_OUT / 16);
    int blocks = (int)((numTiles + 7) / 8);
    gemm_wmma_kernel<<<blocks, 256, 0, stream>>>(transferred, Wm, bias, out, M);
  }
}